// FunctionalGaussianProcess_75634374082815
// MI455X (gfx1250) — compile-verified
//
#include <hip/hip_runtime.h>
#include <hip/hip_bf16.h>

typedef float v2f __attribute__((ext_vector_type(2)));
typedef float v8f __attribute__((ext_vector_type(8)));

#define MT 256   // test items
#define NO 512   // observed items
#define PD 24    // points per item
#define DD 8     // dims per point

// ---------------------------------------------------------------------------
// squared norms of every point: SN[p] = ||x_p||^2
// ---------------------------------------------------------------------------
__global__ __launch_bounds__(256) void sqnorm_kernel(const float* __restrict__ X,
                                                     float* __restrict__ SN, int npts) {
  int p = blockIdx.x * blockDim.x + threadIdx.x;
  if (p < npts) {
    float s = 0.f;
#pragma unroll
    for (int d = 0; d < DD; ++d) { float x = X[p * DD + d]; s += x * x; }
    SN[p] = s;
  }
}

// ---------------------------------------------------------------------------
// self terms: s[i] = sum_{u,v} A[i,u] A[i,v] exp(-0.5*max(d2(x_iu,x_iv),0))
// one wave32 per item i
// ---------------------------------------------------------------------------
__global__ __launch_bounds__(256) void self_kernel(const float* __restrict__ X,
                                                   const float* __restrict__ A,
                                                   const float* __restrict__ SN,
                                                   float* __restrict__ sOut, int n) {
  int wave = threadIdx.x >> 5, lane = threadIdx.x & 31;
  int i = blockIdx.x * (blockDim.x >> 5) + wave;
  float acc = 0.f;
  if (i < n) {
    for (int idx = lane; idx < PD * PD; idx += 32) {
      int u = idx / PD, v = idx % PD;
      const float* xu = X + (i * PD + u) * DD;
      const float* xv = X + (i * PD + v) * DD;
      float dot = 0.f;
#pragma unroll
      for (int d = 0; d < DD; ++d) dot += xu[d] * xv[d];
      float d2 = SN[i * PD + u] + SN[i * PD + v] - 2.f * dot;
      acc += A[i * PD + u] * A[i * PD + v] * __expf(-0.5f * fmaxf(d2, 0.f));
    }
  }
#pragma unroll
  for (int off = 16; off > 0; off >>= 1) acc += __shfl_xor(acc, off);
  if (i < n && lane == 0) sOut[i] = acc;
}

// ---------------------------------------------------------------------------
// Kernel-matrix tile builder.  One wave per 16x16 (i,j) output tile.
// For each (u,v) in 24x24: point-gram S = X1u @ X2v^T via two
// v_wmma_f32_16x16x4_f32 (K = 0..3 and 4..7), then VALU exp + weighted acc.
// C layout: lane l, reg r -> i = i0 + r + 8*(l>>4), j = j0 + (l&15).
// ---------------------------------------------------------------------------
__global__ __launch_bounds__(32) void gram_kernel(
    const float* __restrict__ X1, const float* __restrict__ SN1,
    const float* __restrict__ A1, const float* __restrict__ s1,
    const float* __restrict__ X2, const float* __restrict__ SN2,
    const float* __restrict__ A2, const float* __restrict__ s2,
    float* __restrict__ out, int ldc,
    const float* __restrict__ rho_p, const float* __restrict__ nu_p,
    const float* __restrict__ g_p, int addDiag) {
  const int lane = threadIdx.x & 31;
  const int half = lane >> 4, lq = lane & 15;
  const int i0 = blockIdx.y * 16, j0 = blockIdx.x * 16;

  float acc[8];
#pragma unroll
  for (int r = 0; r < 8; ++r) acc[r] = 0.f;

  for (int u = 0; u < PD; ++u) {
    float a1u[8], n1u[8];
#pragma unroll
    for (int r = 0; r < 8; ++r) {
      int i = i0 + r + 8 * half;
      a1u[r] = A1[i * PD + u];
      n1u[r] = SN1[i * PD + u];
    }
    // A fragment (16x4 f32): lane l holds row lq, dims 2*half + {0,1} (+4 for chunk 1)
    const float* x1 = X1 + ((i0 + lq) * PD + u) * DD;
    v2f a0; a0.x = x1[2 * half + 0]; a0.y = x1[2 * half + 1];
    v2f a1; a1.x = x1[4 + 2 * half + 0]; a1.y = x1[4 + 2 * half + 1];

    for (int v = 0; v < PD; ++v) {
      // B fragment (4x16 f32): lane l holds col lq, dims 2*half + {0,1} (+4)
      const float* x2 = X2 + ((j0 + lq) * PD + v) * DD;
      v2f b0; b0.x = x2[2 * half + 0]; b0.y = x2[2 * half + 1];
      v2f b1; b1.x = x2[4 + 2 * half + 0]; b1.y = x2[4 + 2 * half + 1];

      v8f s = {};
      s = __builtin_amdgcn_wmma_f32_16x16x4_f32(false, a0, false, b0, (short)0, s, false, false);
      s = __builtin_amdgcn_wmma_f32_16x16x4_f32(false, a1, false, b1, (short)0, s, false, false);

      float sn2v = SN2[(j0 + lq) * PD + v];
      float a2v  = A2[(j0 + lq) * PD + v];
#pragma unroll
      for (int r = 0; r < 8; ++r) {
        float d2 = n1u[r] + sn2v - 2.f * s[r];
        acc[r] += a1u[r] * a2v * __expf(-0.5f * fmaxf(d2, 0.f));
      }
    }
  }

  float invrho = 1.f / rho_p[0];
  float nu = nu_p[0], g = g_p[0];
#pragma unroll
  for (int r = 0; r < 8; ++r) {
    int i = i0 + r + 8 * half, j = j0 + lq;
    float d2 = s1[i] + s2[j] - 2.f * acc[r];
    float val = nu * __expf(-0.5f * fmaxf(d2, 0.f) * invrho);
    if (addDiag && i == j) val += nu * g;
    out[i * ldc + j] = val;
  }
}

// ---------------------------------------------------------------------------
// In-place right-looking Cholesky of Koo (512x512), single workgroup.
// Lower triangle holds L afterwards.
// ---------------------------------------------------------------------------
__global__ __launch_bounds__(1024) void chol_kernel(float* __restrict__ Lm) {
  const int n = NO, tid = threadIdx.x, nt = blockDim.x;
  for (int k = 0; k < n; ++k) {
    if (tid == 0) Lm[k * n + k] = sqrtf(Lm[k * n + k]);
    __syncthreads();
    float dinv = 1.f / Lm[k * n + k];
    for (int i = k + 1 + tid; i < n; i += nt) Lm[i * n + k] *= dinv;
    __syncthreads();
    for (int i = k + 1 + tid; i < n; i += nt) {
      float lik = Lm[i * n + k];
      for (int j = k + 1; j <= i; ++j) Lm[i * n + j] -= lik * Lm[j * n + k];
    }
    __syncthreads();
  }
}

// ---------------------------------------------------------------------------
// Forward + backward triangular solves: G[:,col] = Koo^{-1} rhs_col,
// rhs cols 0..255 = Kox columns, col 256 = ones vector.
// One wave32 per column, y staged in LDS, lane-parallel dots.
// G stored column-major: G[col*512 + n].
// ---------------------------------------------------------------------------
#define SW 8
__global__ __launch_bounds__(SW * 32) void solve_kernel(const float* __restrict__ L,
                                                        const float* __restrict__ Kox,
                                                        float* __restrict__ G) {
  __shared__ float y[SW][NO];
  int wave = threadIdx.x >> 5, lane = threadIdx.x & 31;
  int col = blockIdx.x * SW + wave;
  bool valid = col <= MT;  // 0..256
  if (valid)
    for (int n = lane; n < NO; n += 32)
      y[wave][n] = (col < MT) ? Kox[n * MT + col] : 1.0f;
  __syncthreads();
  // forward: L y = b
  for (int r = 0; r < NO; ++r) {
    float s = 0.f;
    if (valid)
      for (int c = lane; c < r; c += 32) s += L[r * NO + c] * y[wave][c];
#pragma unroll
    for (int off = 16; off > 0; off >>= 1) s += __shfl_xor(s, off);
    if (valid && lane == 0) y[wave][r] = (y[wave][r] - s) / L[r * NO + r];
    __syncthreads();
  }
  // backward: L^T x = y
  for (int rr = 0; rr < NO; ++rr) {
    int r = NO - 1 - rr;
    float s = 0.f;
    if (valid)
      for (int c = r + 1 + lane; c < NO; c += 32) s += L[c * NO + r] * y[wave][c];
#pragma unroll
    for (int off = 16; off > 0; off >>= 1) s += __shfl_xor(s, off);
    if (valid && lane == 0) y[wave][r] = (y[wave][r] - s) / L[r * NO + r];
    __syncthreads();
  }
  if (valid)
    for (int n = lane; n < NO; n += 32) G[col * NO + n] = y[wave][n];
}

// ---------------------------------------------------------------------------
// mean row + column sums (Kbx and Koo_inv_sum). One block per column.
// ---------------------------------------------------------------------------
__global__ __launch_bounds__(256) void mean_kernel(const float* __restrict__ G,
                                                   const float* __restrict__ ys,
                                                   const float* __restrict__ b_p,
                                                   float* __restrict__ out,
                                                   float* __restrict__ colsum) {
  __shared__ float r1[256], r2[256];
  int col = blockIdx.x, tid = threadIdx.x;
  float b = b_p[0];
  float sm = 0.f, sc = 0.f;
  for (int n = tid; n < NO; n += 256) {
    float g = G[col * NO + n];
    sm += g * (ys[n] - b);
    sc += g;
  }
  r1[tid] = sm; r2[tid] = sc;
  __syncthreads();
  for (int off = 128; off > 0; off >>= 1) {
    if (tid < off) { r1[tid] += r1[tid + off]; r2[tid] += r2[tid + off]; }
    __syncthreads();
  }
  if (tid == 0) {
    colsum[col] = r2[0];
    if (col < MT) out[col] = b + r1[0];  // mean row (row 0 of output)
  }
}

// ---------------------------------------------------------------------------
// cov = Kxx - gain@Kox + (1-Kbx)^T(1-Kbx)/Koo_inv_sum.
// One wave per 16x16 tile; K=512 via 64 pairs of v_wmma_f32_16x16x4_f32.
// ---------------------------------------------------------------------------
__global__ __launch_bounds__(32) void cov_kernel(const float* __restrict__ Kxx,
                                                 const float* __restrict__ Kox,
                                                 const float* __restrict__ G,
                                                 const float* __restrict__ colsum,
                                                 float* __restrict__ out) {
  const int lane = threadIdx.x & 31;
  const int half = lane >> 4, lq = lane & 15;
  const int i0 = blockIdx.y * 16, j0 = blockIdx.x * 16;
  v8f c = {};
  for (int n0 = 0; n0 < NO; n0 += 8) {
    const float* grow = G + (i0 + lq) * NO + n0 + 2 * half;  // A[m=lq][k]
    v2f a0; a0.x = grow[0]; a0.y = grow[1];
    v2f a1; a1.x = grow[4]; a1.y = grow[5];
    v2f b0; b0.x = Kox[(n0 + 2 * half + 0) * MT + j0 + lq];
            b0.y = Kox[(n0 + 2 * half + 1) * MT + j0 + lq];
    v2f b1; b1.x = Kox[(n0 + 4 + 2 * half + 0) * MT + j0 + lq];
            b1.y = Kox[(n0 + 4 + 2 * half + 1) * MT + j0 + lq];
    c = __builtin_amdgcn_wmma_f32_16x16x4_f32(false, a0, false, b0, (short)0, c, false, false);
    c = __builtin_amdgcn_wmma_f32_16x16x4_f32(false, a1, false, b1, (short)0, c, false, false);
  }
  float invS = 1.f / colsum[MT];       // Koo_inv_sum is column 256
  float kbxj = colsum[j0 + lq];
#pragma unroll
  for (int r = 0; r < 8; ++r) {
    int i = i0 + r + 8 * half, j = j0 + lq;
    float v = Kxx[i * MT + j] - c[r] + (1.f - colsum[i]) * (1.f - kbxj) * invS;
    out[(1 + i) * MT + j] = v;
  }
}

// ---------------------------------------------------------------------------
extern "C" void kernel_launch(void* const* d_in, const int* in_sizes, int n_in,
                              void* d_out, int out_size, void* d_ws, size_t ws_size,
                              hipStream_t stream) {
  const float* Xt  = (const float*)d_in[0];  // X_test  [256,24,8]
  const float* At  = (const float*)d_in[1];  // A_test  [256,24]
  const float* Xo  = (const float*)d_in[2];  // X_obs   [512,24,8]
  const float* Ao  = (const float*)d_in[3];  // A_obs   [512,24]
  const float* ys  = (const float*)d_in[4];  // observed_ys [512]
  const float* rho = (const float*)d_in[5];
  const float* g   = (const float*)d_in[6];
  const float* nu  = (const float*)d_in[7];
  const float* b   = (const float*)d_in[8];
  float* out = (float*)d_out;

  // workspace layout (floats)
  float* ws  = (float*)d_ws;
  float* SN1 = ws;                      // 256*24  = 6144
  float* SN2 = SN1 + MT * PD;           // 512*24  = 12288
  float* sT  = SN2 + NO * PD;           // 256
  float* sO  = sT + MT;                 // 512
  float* Kxx = sO + NO;                 // 65536
  float* Kox = Kxx + MT * MT;           // 131072
  float* Koo = Kox + NO * MT;           // 262144
  float* G   = Koo + NO * NO;           // 257*512 = 131584
  float* csm = G + (MT + 1) * NO;       // 257

  sqnorm_kernel<<<(MT * PD + 255) / 256, 256, 0, stream>>>(Xt, SN1, MT * PD);
  sqnorm_kernel<<<(NO * PD + 255) / 256, 256, 0, stream>>>(Xo, SN2, NO * PD);
  self_kernel<<<(MT + 7) / 8, 256, 0, stream>>>(Xt, At, SN1, sT, MT);
  self_kernel<<<(NO + 7) / 8, 256, 0, stream>>>(Xo, Ao, SN2, sO, NO);

  gram_kernel<<<dim3(16, 16), 32, 0, stream>>>(Xt, SN1, At, sT, Xt, SN1, At, sT,
                                               Kxx, MT, rho, nu, g, 0);
  gram_kernel<<<dim3(16, 32), 32, 0, stream>>>(Xo, SN2, Ao, sO, Xt, SN1, At, sT,
                                               Kox, MT, rho, nu, g, 0);
  gram_kernel<<<dim3(32, 32), 32, 0, stream>>>(Xo, SN2, Ao, sO, Xo, SN2, Ao, sO,
                                               Koo, NO, rho, nu, g, 1);

  chol_kernel<<<1, 1024, 0, stream>>>(Koo);
  solve_kernel<<<(MT + 1 + SW - 1) / SW, SW * 32, 0, stream>>>(Koo, Kox, G);
  mean_kernel<<<MT + 1, 256, 0, stream>>>(G, ys, b, out, csm);
  cov_kernel<<<dim3(16, 16), 32, 0, stream>>>(Kxx, Kox, G, csm, out);
}